// MaxChannelPool_57080115364017
// MI455X (gfx1250) — compile-verified
//
#include <hip/hip_runtime.h>

// MaxChannelPool for x:(8,512,64,64) f32 -> out:(8,256,64,64) f32
// Reference quirk: window = KERNEL_SIZE-1 = 6, stride 2, zero-pad 3:
//   out[n,o,s] = max_{k=0..5} xp[n, 2o+k, s]
// Pairwise-max factorization: p_j = max(xp[2j], xp[2j+1]);
//   out[o] = max(p_o, p_{o+1}, p_{o+2})
// => each input element loaded exactly once per computing thread.
//
// Roofline: 64 MiB read + 32 MiB write ~ 101 MB -> ~4.3 us @ 23.3 TB/s.
// Pure bandwidth kernel; WMMA inapplicable (max-reduce is not a contraction).
//
// Mapping (reference shapes hardcoded):
//   C=512, CO=256, S=4096 (=> 1024 float4/plane), L=8 outputs/thread.
//   Block = 256 threads owning 256 consecutive float4 spatial positions of
//   ONE (n, o-tile): n/ot/j0 are blockIdx-derived SGPR values => all pad
//   branches are scalar, loads use one base + immediate offsets.
//   Grid = N * 32 o-tiles * 4 spatial blocks = 1024 blocks (8192 wave32s).

typedef float f4 __attribute__((ext_vector_type(4)));

#define C_IN   512
#define S4_LOG 10          // 1024 float4 per channel plane
#define L_OUT  8
#define OT_CNT 32          // 256 / L_OUT

__device__ __forceinline__ f4 vmax4(f4 a, f4 b) {
    f4 r;
    r.x = fmaxf(a.x, b.x);
    r.y = fmaxf(a.y, b.y);
    r.z = fmaxf(a.z, b.z);
    r.w = fmaxf(a.w, b.w);
    return r;
}

// p_j = max(xp[2j], xp[2j+1]); real channels a = 2j-3, b = 2j-2 (OOB -> 0).
// j is block-uniform (SGPR) => these branches are scalar s_cbranch.
__device__ __forceinline__ f4 pair_max(const f4* __restrict__ xc, int j) {
    const int a = 2 * j - 3;
    const int b = 2 * j - 2;
    const f4 z = {0.0f, 0.0f, 0.0f, 0.0f};
    if (a >= 0) {                                   // a <= 511 by construction
        f4 va = xc[(size_t)a << S4_LOG];
        if (b < C_IN) {                             // interior: common path
            f4 vb = xc[(size_t)b << S4_LOG];
            return vmax4(va, vb);
        }
        return vmax4(va, z);                        // b == 512 pad (j == 257)
    }
    if (b >= 0) {                                   // j == 1: a pad, b == 0
        return vmax4(xc[(size_t)b << S4_LOG], z);
    }
    return z;                                       // j == 0: both pads
}

__global__ void __launch_bounds__(256)
maxpool_chan_kernel(const f4* __restrict__ x, f4* __restrict__ y, int nblocks) {
    const int b = blockIdx.x;
    if (b >= nblocks) return;                       // scalar guard (grid exact)

    // Block-uniform (SGPR) decomposition: 4 spatial blocks per (n, o-tile).
    const int sblk = b & 3;
    const int ot   = (b >> 2) & (OT_CNT - 1);
    const int n    = b >> 7;

    const int s4 = (sblk << 8) + (int)threadIdx.x;  // lane-varying, coalesced

    // n*C*S4 = n<<19 ; n*CO*S4 = n<<18 (float4 units)
    const f4* __restrict__ xc = x + ((size_t)n << 19) + (size_t)s4;
    f4* __restrict__ yc       = y + ((size_t)n << 18)
                                  + ((size_t)(ot * L_OUT) << S4_LOG) + (size_t)s4;

    const int j0 = ot * L_OUT;                      // SGPR
    f4 pm2 = pair_max(xc, j0);                      // p_{j0}
    f4 pm1 = pair_max(xc, j0 + 1);                  // p_{j0+1}

#pragma unroll
    for (int t = 0; t < L_OUT; ++t) {
        const int j = j0 + 2 + t;

        // Warm L2 ~6 pairs ahead: 16 KiB channel stride defeats HW stride
        // prefetch; global_prefetch_b8 doesn't touch LOADcnt.
        int pf = 2 * j + 9;
        if (pf > C_IN - 1) pf = C_IN - 1;
        __builtin_prefetch((const void*)(xc + ((size_t)pf << S4_LOG)), 0, 3);

        const f4 p = pair_max(xc, j);               // 2 x global_load_b128
        const f4 r = vmax4(vmax4(pm2, pm1), p);     // out[o = j-2]

        // Output is write-once / never re-read: NT store keeps the 64 MiB
        // input resident in the 192 MB L2.
        __builtin_nontemporal_store(r, yc + ((size_t)t << S4_LOG));

        pm2 = pm1;
        pm1 = p;
    }
}

extern "C" void kernel_launch(void* const* d_in, const int* in_sizes, int n_in,
                              void* d_out, int out_size, void* d_ws, size_t ws_size,
                              hipStream_t stream) {
    (void)n_in; (void)out_size; (void)d_ws; (void)ws_size;

    const f4* x = (const f4*)d_in[0];
    f4* y = (f4*)d_out;

    const int N = in_sizes[0] / (C_IN * 64 * 64);   // 8 for the reference
    const int nblocks = N * OT_CNT * 4;             // 1024 for N=8

    maxpool_chan_kernel<<<dim3(nblocks), dim3(256), 0, stream>>>(x, y, nblocks);
}